// ppo_policy_46454366273847
// MI455X (gfx1250) — compile-verified
//
#include <hip/hip_runtime.h>
#include <hip/hip_bf16.h>
#include <math.h>

// ---------------------------------------------------------------------------
// GatedGraphConv x3 (+GRU, L=10) + pooling + MLP heads for MI455X (gfx1250).
//
// Strategy (see analysis): the workload is dominated by edge gather/scatter
// (L2-resident on a 192MB L2), the GEMMs are folded into per-16-node-tile
// V_WMMA_F32_16X16X4_F32 chains (full fp32 precision), and the W_l edge
// transform is algebraically fused into the GRU input weights
// (M_l = W_l @ Wih^T) to delete one GEMM + 25MB of traffic per step.
// ---------------------------------------------------------------------------

#define N_NODES_C 100000
#define N_EDGES_C 1600000
#define N_GRAPHS_C 1024
#define D_C 32

typedef float v2f __attribute__((ext_vector_type(2)));
typedef float v8f __attribute__((ext_vector_type(8)));

__device__ __forceinline__ float sigmoidf_(float x) { return 1.0f / (1.0f + expf(-x)); }

// -------------------------------------------------------------------------
// Zero a float buffer (grid-stride).
// -------------------------------------------------------------------------
__global__ void zero_kernel(float* __restrict__ p, int n) {
    int i = blockIdx.x * blockDim.x + threadIdx.x;
    int stride = gridDim.x * blockDim.x;
    for (; i < n; i += stride) p[i] = 0.0f;
}

// -------------------------------------------------------------------------
// Precompute fused input weights: M_l[k][c] = sum_j W_l[k][j] * Wih[c][j]
// (so gi = segment_sum(h[src]) @ M_l).  W: [10,32,32], Wih: [96,32],
// Mf: [10,32,96].  Tiny; one block per step l.
// -------------------------------------------------------------------------
__global__ void fuse_kernel(const float* __restrict__ W,
                            const float* __restrict__ Wih,
                            float* __restrict__ Mf) {
    int l = blockIdx.x;
    const float* Wl = W + l * 32 * 32;
    float* M = Mf + l * 32 * 96;
    for (int idx = threadIdx.x; idx < 32 * 96; idx += blockDim.x) {
        int k = idx / 96, c = idx % 96;
        float s = 0.0f;
        #pragma unroll
        for (int j = 0; j < 32; ++j) s += Wl[k * 32 + j] * Wih[c * 32 + j];
        M[idx] = s;
    }
}

// -------------------------------------------------------------------------
// Edge scatter: agg[dst] += h[src].  8 lanes per edge, each lane moves a
// float4 (global_load_b128 + 4x global_atomic_add_f32).  Both h and agg are
// 12.8MB -> L2-resident; atomics resolve in L2.
// -------------------------------------------------------------------------
__global__ void scatter_kernel(const int* __restrict__ ei,
                               const float* __restrict__ h,
                               float* __restrict__ agg, int n_edges) {
    int tid = blockIdx.x * blockDim.x + threadIdx.x;
    if (tid >= n_edges * 8) return;
    int e = tid >> 3;
    int part = tid & 7;
    int src = ei[e];
    int dst = ei[n_edges + e];
    const float4 v = *reinterpret_cast<const float4*>(h + (size_t)src * 32 + part * 4);
    float* p = agg + (size_t)dst * 32 + part * 4;
    atomicAdd(p + 0, v.x);
    atomicAdd(p + 1, v.y);
    atomicAdd(p + 2, v.z);
    atomicAdd(p + 3, v.w);
}

// -------------------------------------------------------------------------
// GRU step: one wave owns one 16-node tile.
//   gi = agg_raw @ M_l + bih          (16x32 @ 32x96)
//   gh = h       @ Whh^T + bhh        (16x32 @ 32x96)
//   r = sig(gi_r+gh_r); z = sig(gi_z+gh_z); n = tanh(gi_n + r*gh_n)
//   h' = (1-z)*n + z*h   (ReLU on final step of each conv)
// GEMMs use V_WMMA_F32_16X16X4_F32: 6 N-tiles x 8 K-steps x 2 matrices.
// Fragment mapping (per ISA 7.12.2, f32 16x16x4):
//   A: lane&15 = row M, lane>>4 selects K-pair {0,1}/{2,3}; v0=K even, v1=K odd
//   B: lane&15 = col N, same K-pair split
//   C/D: lane&15 = col N, VGPR i = row (i + 8*(lane>>4))
// Early tile-exit is wave-uniform so EXEC is all-ones at every WMMA.
// -------------------------------------------------------------------------
__global__ __launch_bounds__(256) void gru_step_kernel(
    const float* __restrict__ agg,   // [N,32] raw neighbor sums
    const float* __restrict__ hin,   // [N,32]
    float* __restrict__ hout,        // [N,32]
    const float* __restrict__ Mf,    // [32,96] fused W_l @ Wih^T
    const float* __restrict__ Whh,   // [96,32]
    const float* __restrict__ bih,   // [96]
    const float* __restrict__ bhh,   // [96]
    int n_nodes, int apply_relu) {
    __shared__ float sM[32 * 96];    // M_l, row-major [k][c]
    __shared__ float sW[96 * 32];    // Whh, row-major [c][k]
    for (int i = threadIdx.x; i < 32 * 96; i += blockDim.x) {
        sM[i] = Mf[i];
        sW[i] = Whh[i];
    }
    __syncthreads();

    const int wave = threadIdx.x >> 5;           // 8 waves / block
    const int lane = threadIdx.x & 31;
    const int tile = blockIdx.x * 8 + wave;
    if (tile * 16 >= n_nodes) return;            // wave-uniform
    const int nbase = tile * 16;

    const int col = lane & 15;
    const int half = lane >> 4;
    const int kc = half * 2;

    // A fragments for both GEMMs (row = lane&15, 8 K-steps of 4).
    v2f aA[8], aH[8];
    #pragma unroll
    for (int s = 0; s < 8; ++s) {
        int k = s * 4 + kc;
        const float* pa = agg + (size_t)(nbase + col) * 32 + k;
        const float* ph = hin + (size_t)(nbase + col) * 32 + k;
        v2f ta; ta.x = pa[0]; ta.y = pa[1]; aA[s] = ta;
        v2f th; th.x = ph[0]; th.y = ph[1]; aH[s] = th;
    }

    // r/z tiles (0..3): gi and gh accumulate into the same register tile.
    v8f accRZ[4];
    #pragma unroll
    for (int nt = 0; nt < 4; ++nt) {
        v8f c = {};
        #pragma unroll
        for (int s = 0; s < 8; ++s) {
            int k = s * 4 + kc;
            v2f b;
            b.x = sM[(k + 0) * 96 + nt * 16 + col];
            b.y = sM[(k + 1) * 96 + nt * 16 + col];
            c = __builtin_amdgcn_wmma_f32_16x16x4_f32(false, aA[s], false, b,
                                                      (short)0, c, false, false);
        }
        #pragma unroll
        for (int s = 0; s < 8; ++s) {
            int k = s * 4 + kc;
            v2f b;
            b.x = sW[(nt * 16 + col) * 32 + k + 0];
            b.y = sW[(nt * 16 + col) * 32 + k + 1];
            c = __builtin_amdgcn_wmma_f32_16x16x4_f32(false, aH[s], false, b,
                                                      (short)0, c, false, false);
        }
        accRZ[nt] = c;
    }

    // n tiles (4,5): keep gi and gh separate (n = tanh(i_n + r*h_n)).
    v8f accIN[2], accHN[2];
    #pragma unroll
    for (int t = 0; t < 2; ++t) {
        int nt = 4 + t;
        v8f ci = {};
        v8f ch = {};
        #pragma unroll
        for (int s = 0; s < 8; ++s) {
            int k = s * 4 + kc;
            v2f b;
            b.x = sM[(k + 0) * 96 + nt * 16 + col];
            b.y = sM[(k + 1) * 96 + nt * 16 + col];
            ci = __builtin_amdgcn_wmma_f32_16x16x4_f32(false, aA[s], false, b,
                                                       (short)0, ci, false, false);
        }
        #pragma unroll
        for (int s = 0; s < 8; ++s) {
            int k = s * 4 + kc;
            v2f b;
            b.x = sW[(nt * 16 + col) * 32 + k + 0];
            b.y = sW[(nt * 16 + col) * 32 + k + 1];
            ch = __builtin_amdgcn_wmma_f32_16x16x4_f32(false, aH[s], false, b,
                                                       (short)0, ch, false, false);
        }
        accIN[t] = ci;
        accHN[t] = ch;
    }

    // Column-constant biases for this lane's gate columns.
    float brz[4], bi_n[2], bh_n[2];
    #pragma unroll
    for (int nt = 0; nt < 4; ++nt) brz[nt] = bih[nt * 16 + col] + bhh[nt * 16 + col];
    #pragma unroll
    for (int t = 0; t < 2; ++t) {
        bi_n[t] = bih[64 + t * 16 + col];
        bh_n[t] = bhh[64 + t * 16 + col];
    }

    // Gates + blend + store.  Lane owns node-column c = t*16 + col, rows
    // rrow = i + 8*half for VGPR i.
    #pragma unroll
    for (int t = 0; t < 2; ++t) {
        int c = t * 16 + col;
        #pragma unroll
        for (int i = 0; i < 8; ++i) {
            int rrow = i + half * 8;
            float r = sigmoidf_(accRZ[t][i] + brz[t]);
            float z = sigmoidf_(accRZ[t + 2][i] + brz[t + 2]);
            float nn = tanhf(accIN[t][i] + bi_n[t] + r * (accHN[t][i] + bh_n[t]));
            float hv = hin[(size_t)(nbase + rrow) * 32 + c];
            float hn = (1.0f - z) * nn + z * hv;
            if (apply_relu) hn = fmaxf(hn, 0.0f);
            hout[(size_t)(nbase + rrow) * 32 + c] = hn;
        }
    }
}

// -------------------------------------------------------------------------
// Graph pooling: sums[g][d] += h[n][d]; counts[g] += 1.  32 lanes per node.
// -------------------------------------------------------------------------
__global__ void pool_kernel(const float* __restrict__ h,
                            const int* __restrict__ batch,
                            float* __restrict__ sums,
                            float* __restrict__ counts, int n_nodes) {
    int tid = blockIdx.x * blockDim.x + threadIdx.x;
    if (tid >= n_nodes * 32) return;
    int n = tid >> 5;
    int d = tid & 31;
    int g = batch[n];
    atomicAdd(&sums[(size_t)g * 32 + d], h[(size_t)n * 32 + d]);
    if (d == 0) atomicAdd(&counts[g], 1.0f);
}

// -------------------------------------------------------------------------
// Per-graph heads (t and b outputs).  1024 graphs, ~3.4k MACs each: trivial.
// -------------------------------------------------------------------------
__global__ void graph_head_kernel(
    const float* __restrict__ sums, const float* __restrict__ counts,
    const float* __restrict__ g1w, const float* __restrict__ g1b,
    const float* __restrict__ g2w, const float* __restrict__ g2b,
    const float* __restrict__ t1w, const float* __restrict__ t1b,
    const float* __restrict__ t2w, const float* __restrict__ t2b,
    const float* __restrict__ b1w, const float* __restrict__ b1b,
    const float* __restrict__ b2w, const float* __restrict__ b2b,
    float* __restrict__ out_t, float* __restrict__ out_b, int n_graphs) {
    int g = blockIdx.x * blockDim.x + threadIdx.x;
    if (g >= n_graphs) return;
    float s[32], gm[32];
    float cnt = fmaxf(counts[g], 1.0f);
    #pragma unroll
    for (int d = 0; d < 32; ++d) {
        s[d] = sums[(size_t)g * 32 + d];
        gm[d] = s[d] / cnt;
    }
    float a24[24];
    #pragma unroll
    for (int o = 0; o < 24; ++o) {
        float acc = g1b[o];
        #pragma unroll
        for (int i = 0; i < 32; ++i) acc += g1w[o * 32 + i] * gm[i];
        a24[o] = acc;                                   // no activation
    }
    float gv[32];
    #pragma unroll
    for (int o = 0; o < 32; ++o) {
        float acc = g2b[o];
        #pragma unroll
        for (int i = 0; i < 24; ++i) acc += g2w[o * 24 + i] * a24[i];
        gv[o] = acc;
    }
    float th[32];
    #pragma unroll
    for (int o = 0; o < 32; ++o) {
        float acc = t1b[o];
        #pragma unroll
        for (int i = 0; i < 32; ++i) acc += t1w[o * 32 + i] * gv[i];
        th[o] = fmaxf(acc, 0.0f);
    }
    #pragma unroll
    for (int o = 0; o < 2; ++o) {
        float acc = t2b[o];
        #pragma unroll
        for (int i = 0; i < 32; ++i) acc += t2w[o * 32 + i] * th[i];
        out_t[g * 2 + o] = fmaxf(acc, 0.0f);
    }
    float bh[16];
    #pragma unroll
    for (int o = 0; o < 16; ++o) {
        float acc = b1b[o];
        #pragma unroll
        for (int i = 0; i < 32; ++i) acc += b1w[o * 32 + i] * s[i];  // add-pool = sums
        bh[o] = fmaxf(acc, 0.0f);
    }
    #pragma unroll
    for (int o = 0; o < 3; ++o) {
        float acc = b2b[o];
        #pragma unroll
        for (int i = 0; i < 16; ++i) acc += b2w[o * 16 + i] * bh[i];
        out_b[g * 3 + o] = fmaxf(acc, 0.0f);
    }
}

// -------------------------------------------------------------------------
// Per-node head: relu(n2 @ relu(n1 @ h + b1) + b2), scalar output.
// -------------------------------------------------------------------------
__global__ void node_head_kernel(const float* __restrict__ h,
                                 const float* __restrict__ n1w,
                                 const float* __restrict__ n1b,
                                 const float* __restrict__ n2w,
                                 const float* __restrict__ n2b,
                                 float* __restrict__ out_n, int n_nodes) {
    int n = blockIdx.x * blockDim.x + threadIdx.x;
    if (n >= n_nodes) return;
    float hv[32];
    const float4* hp = reinterpret_cast<const float4*>(h + (size_t)n * 32);
    #pragma unroll
    for (int q = 0; q < 8; ++q) {
        float4 v = hp[q];
        hv[q * 4 + 0] = v.x; hv[q * 4 + 1] = v.y;
        hv[q * 4 + 2] = v.z; hv[q * 4 + 3] = v.w;
    }
    float acc2 = n2b[0];
    #pragma unroll
    for (int o = 0; o < 16; ++o) {
        float a = n1b[o];
        #pragma unroll
        for (int i = 0; i < 32; ++i) a += n1w[o * 32 + i] * hv[i];
        acc2 += n2w[o] * fmaxf(a, 0.0f);
    }
    out_n[n] = fmaxf(acc2, 0.0f);
}

// -------------------------------------------------------------------------
// Host launch.  Input order follows setup_inputs() dict order with params
// flattened jax-pytree style (sorted dict keys):
//  0:x 1:edge_index 2:batch 3:b1.b 4:b1.w 5:b2.b 6:b2.w
//  7..21: conv{1,2,3} x {W, Whh, Wih, bhh, bih}
//  22:g1.b 23:g1.w 24:g2.b 25:g2.w 26:n1.b 27:n1.w 28:n2.b 29:n2.w
//  30:t1.b 31:t1.w 32:t2.b 33:t2.w
// -------------------------------------------------------------------------
extern "C" void kernel_launch(void* const* d_in, const int* in_sizes, int n_in,
                              void* d_out, int out_size, void* d_ws, size_t ws_size,
                              hipStream_t stream) {
    const float* x     = (const float*)d_in[0];
    const int*   ei    = (const int*)d_in[1];
    const int*   batch = (const int*)d_in[2];
    const float* b1b = (const float*)d_in[3];
    const float* b1w = (const float*)d_in[4];
    const float* b2b = (const float*)d_in[5];
    const float* b2w = (const float*)d_in[6];
    const float *convW[3], *convWhh[3], *convWih[3], *convBhh[3], *convBih[3];
    for (int c = 0; c < 3; ++c) {
        convW[c]   = (const float*)d_in[7 + c * 5 + 0];
        convWhh[c] = (const float*)d_in[7 + c * 5 + 1];
        convWih[c] = (const float*)d_in[7 + c * 5 + 2];
        convBhh[c] = (const float*)d_in[7 + c * 5 + 3];
        convBih[c] = (const float*)d_in[7 + c * 5 + 4];
    }
    const float* g1b = (const float*)d_in[22];
    const float* g1w = (const float*)d_in[23];
    const float* g2b = (const float*)d_in[24];
    const float* g2w = (const float*)d_in[25];
    const float* n1b = (const float*)d_in[26];
    const float* n1w = (const float*)d_in[27];
    const float* n2b = (const float*)d_in[28];
    const float* n2w = (const float*)d_in[29];
    const float* t1b = (const float*)d_in[30];
    const float* t1w = (const float*)d_in[31];
    const float* t2b = (const float*)d_in[32];
    const float* t2w = (const float*)d_in[33];

    // Workspace layout (floats): ~38.9 MB total.
    float* ws     = (float*)d_ws;
    float* hA     = ws;                        // 3,200,000
    float* hB     = hA + (size_t)N_NODES_C * 32;
    float* agg    = hB + (size_t)N_NODES_C * 32;
    float* Mf     = agg + (size_t)N_NODES_C * 32;   // 3*10*32*96 = 92,160
    float* sums   = Mf + 3 * 10 * 32 * 96;          // 32,768
    float* counts = sums + (size_t)N_GRAPHS_C * 32; // 1,024

    // Fuse W_l into GRU input weights once (30 tiny 32x96 matrices).
    for (int c = 0; c < 3; ++c)
        fuse_kernel<<<10, 128, 0, stream>>>(convW[c], convWih[c],
                                            Mf + (size_t)c * 10 * 32 * 96);

    const float* hcur = x;      // never mutate the input
    float* hnxt = hA;
    const int scatter_threads = N_EDGES_C * 8;
    const int gru_blocks = (N_NODES_C / 16 + 7) / 8;
    for (int c = 0; c < 3; ++c) {
        for (int l = 0; l < 10; ++l) {
            zero_kernel<<<2048, 256, 0, stream>>>(agg, N_NODES_C * 32);
            scatter_kernel<<<(scatter_threads + 255) / 256, 256, 0, stream>>>(
                ei, hcur, agg, N_EDGES_C);
            gru_step_kernel<<<gru_blocks, 256, 0, stream>>>(
                agg, hcur, hnxt, Mf + (size_t)(c * 10 + l) * 32 * 96,
                convWhh[c], convBih[c], convBhh[c], N_NODES_C, (l == 9) ? 1 : 0);
            hcur = hnxt;
            hnxt = (hcur == hA) ? hB : hA;
        }
    }

    // Pooling.
    zero_kernel<<<256, 256, 0, stream>>>(sums, N_GRAPHS_C * 32 + N_GRAPHS_C);
    pool_kernel<<<(N_NODES_C * 32 + 255) / 256, 256, 0, stream>>>(
        hcur, batch, sums, counts, N_NODES_C);

    // Outputs: t [1024,2] | n [100000,1] | b [1024,3], concatenated flat.
    float* out   = (float*)d_out;
    float* out_t = out;
    float* out_n = out + N_GRAPHS_C * 2;
    float* out_b = out + N_GRAPHS_C * 2 + N_NODES_C;

    graph_head_kernel<<<(N_GRAPHS_C + 255) / 256, 256, 0, stream>>>(
        sums, counts, g1w, g1b, g2w, g2b, t1w, t1b, t2w, t2b,
        b1w, b1b, b2w, b2b, out_t, out_b, N_GRAPHS_C);
    node_head_kernel<<<(N_NODES_C + 255) / 256, 256, 0, stream>>>(
        hcur, n1w, n1b, n2w, n2b, out_n, N_NODES_C);
}